// WeightedSmoothedComplex_46213848105403
// MI455X (gfx1250) — compile-verified
//
#include <hip/hip_runtime.h>
#include <math.h>

// CDNA5 / gfx1250: wave32, WMMA f32 16x16x4.
typedef __attribute__((ext_vector_type(2))) float v2f;
typedef __attribute__((ext_vector_type(8))) float v8f;

#define W_IN   512
#define H_IN   512
#define WO     449
#define HO     449
#define NIMG   64              // 32 batch * 2 channels
#define IMG_IN (W_IN * H_IN)   // 262144
#define IMG_T1 (H_IN * WO)     // 512*449
#define IMG_S  (HO * WO)       // 449*449
#define PLANE  (HO * WO)       // 201601
#define N_MEAN 6451232.0f      // 32*449*449

__device__ __forceinline__ float band64(int lo, int p) {
  // 1.0 iff lo <= p <= lo+63  (64-tap box window)
  return (p >= lo && p <= lo + 63) ? 1.0f : 0.0f;
}

// ---------------------------------------------------------------------------
// Stage 1: horizontal 64-tap box sum via WMMA.
// out_h[r,c] = sum_{k=0..63} x[r, c+k],  r in [0,512), c in [0,449)
// Per wave: one 16x16 tile. A = input block (M=rows, K=cols), B = band S^T.
// ---------------------------------------------------------------------------
__global__ __launch_bounds__(256)
void hpass_wmma(const float* __restrict__ x, float* __restrict__ t1) {
  const int lane = threadIdx.x & 31;
  const int wid  = blockIdx.x * 8 + (threadIdx.x >> 5);   // 8 waves / block
  const int img  = wid / (32 * 29);
  const int rem  = wid % (32 * 29);
  const int r0   = (rem / 29) << 4;
  const int c0   = (rem % 29) << 4;
  const int hi   = lane >> 4;   // lane half: selects K = {0,1} vs {2,3}
  const int ln   = lane & 15;   // M for A rows, N for B cols

  const float* xi = x + (size_t)img * IMG_IN + (size_t)(r0 + ln) * W_IN;

  v8f acc = {0.f, 0.f, 0.f, 0.f, 0.f, 0.f, 0.f, 0.f};
#pragma unroll
  for (int cch = 0; cch < 20; ++cch) {        // K = 80 (p = 0..79)
    const int p0 = (cch << 2) + (hi << 1);
    int col0 = c0 + p0;
    int col1 = col0 + 1;
    col0 = col0 > (W_IN - 1) ? (W_IN - 1) : col0;   // clamp; band is 0 there
    col1 = col1 > (W_IN - 1) ? (W_IN - 1) : col1;
    v2f a; a.x = xi[col0];          a.y = xi[col1];          // A: input
    v2f b; b.x = band64(ln, p0);    b.y = band64(ln, p0 + 1); // B: band S^T
    acc = __builtin_amdgcn_wmma_f32_16x16x4_f32(
        false, a, false, b, (short)0, acc, false, false);
  }

  const int ccol = c0 + ln;                   // D col = lane&15
  if (ccol < WO) {
    float* to = t1 + (size_t)img * IMG_T1 + ccol;
#pragma unroll
    for (int v = 0; v < 8; ++v) {             // D row = v + 8*half
      const int rr = r0 + v + (hi << 3);
      to[(size_t)rr * WO] = acc[v];
    }
  }
}

// ---------------------------------------------------------------------------
// Stage 2: vertical 64-tap box sum via WMMA + 1/4096 scale.
// s[r,c] = (1/4096) * sum_{k=0..63} t1[r+k, c],  r,c in [0,449)
// Per wave: one 16x16 tile. A = band S, B = t1 row block.
// ---------------------------------------------------------------------------
__global__ __launch_bounds__(256)
void vpass_wmma(const float* __restrict__ t1, float* __restrict__ s) {
  const int lane = threadIdx.x & 31;
  const int wid  = blockIdx.x * 8 + (threadIdx.x >> 5);
  const int img  = wid / (29 * 29);
  const int rem  = wid % (29 * 29);
  const int r0   = (rem / 29) << 4;
  const int c0   = (rem % 29) << 4;
  const int hi   = lane >> 4;
  const int ln   = lane & 15;

  int bcol = c0 + ln;
  bcol = bcol > (WO - 1) ? (WO - 1) : bcol;   // clamp; store masked below
  const float* ti = t1 + (size_t)img * IMG_T1 + bcol;

  v8f acc = {0.f, 0.f, 0.f, 0.f, 0.f, 0.f, 0.f, 0.f};
#pragma unroll
  for (int cch = 0; cch < 20; ++cch) {
    const int p0 = (cch << 2) + (hi << 1);
    int row0 = r0 + p0;
    int row1 = row0 + 1;
    row0 = row0 > (H_IN - 1) ? (H_IN - 1) : row0;   // clamp; band is 0 there
    row1 = row1 > (H_IN - 1) ? (H_IN - 1) : row1;
    v2f a; a.x = band64(ln, p0);         a.y = band64(ln, p0 + 1); // A: band S
    v2f b; b.x = ti[(size_t)row0 * WO];  b.y = ti[(size_t)row1 * WO];
    acc = __builtin_amdgcn_wmma_f32_16x16x4_f32(
        false, a, false, b, (short)0, acc, false, false);
  }

  const float scale = 1.0f / 4096.0f;  // exact power of two
  const int ccol = c0 + ln;
  if (ccol < WO) {
    float* so = s + (size_t)img * IMG_S + ccol;
#pragma unroll
    for (int v = 0; v < 8; ++v) {
      const int rr = r0 + v + (hi << 3);
      if (rr < HO) so[(size_t)rr * WO] = acc[v] * scale;
    }
  }
}

// ---------------------------------------------------------------------------
// Stage 3: global max of |z|^2 for both smoothed tensors.
// Bitwise atomicMax on unsigned is order-preserving for non-negative floats.
// ---------------------------------------------------------------------------
__global__ void init_kernel(unsigned* __restrict__ maxslots) {
  if (threadIdx.x < 2) maxslots[threadIdx.x] = 0u;  // 0u == 0.0f bits
}

__global__ __launch_bounds__(256)
void max_kernel(const float* __restrict__ sp, const float* __restrict__ st,
                unsigned* __restrict__ maxslots) {
  __shared__ float smp[256];
  __shared__ float smt[256];
  float mp = 0.f, mt = 0.f;
  const long long N = 32LL * PLANE;
  for (long long i = (long long)blockIdx.x * 256 + threadIdx.x; i < N;
       i += (long long)gridDim.x * 256) {
    const int b = (int)(i / PLANE);
    const int r = (int)(i % PLANE);
    const size_t o0 = (size_t)b * 2 * PLANE + r;
    const size_t o1 = o0 + PLANE;
    const float p0 = sp[o0], p1 = sp[o1];
    mp = fmaxf(mp, p0 * p0 + p1 * p1);
    const float t0 = st[o0], t1v = st[o1];
    mt = fmaxf(mt, t0 * t0 + t1v * t1v);
  }
  smp[threadIdx.x] = mp;
  smt[threadIdx.x] = mt;
  __syncthreads();
  for (int sft = 128; sft > 0; sft >>= 1) {
    if ((int)threadIdx.x < sft) {
      smp[threadIdx.x] = fmaxf(smp[threadIdx.x], smp[threadIdx.x + sft]);
      smt[threadIdx.x] = fmaxf(smt[threadIdx.x], smt[threadIdx.x + sft]);
    }
    __syncthreads();
  }
  if (threadIdx.x == 0) {
    atomicMax(&maxslots[0], __float_as_uint(smp[0]));
    atomicMax(&maxslots[1], __float_as_uint(smt[0]));
  }
}

// ---------------------------------------------------------------------------
// Stage 4: weighted angles + |diff|, fixed-shape block partial sums.
// ---------------------------------------------------------------------------
__global__ __launch_bounds__(256)
void loss_kernel(const float* __restrict__ sp, const float* __restrict__ st,
                 const unsigned* __restrict__ maxslots,
                 float* __restrict__ partials) {
  __shared__ float sm[256];
  const float invp = 1.0f / __uint_as_float(maxslots[0]);
  const float invt = 1.0f / __uint_as_float(maxslots[1]);
  float acc = 0.f;
  const long long N = 32LL * PLANE;
  for (long long i = (long long)blockIdx.x * 256 + threadIdx.x; i < N;
       i += (long long)gridDim.x * 256) {
    const int b = (int)(i / PLANE);
    const int r = (int)(i % PLANE);
    const size_t o0 = (size_t)b * 2 * PLANE + r;
    const size_t o1 = o0 + PLANE;
    const float p0 = sp[o0], p1 = sp[o1];
    const float t0 = st[o0], t1v = st[o1];
    const float wp = (p0 * p0 + p1 * p1) * invp;
    const float wt = (t0 * t0 + t1v * t1v) * invt;
    const float yp = atan2f(p1, p0) * wp * wp;
    const float yt = atan2f(t1v, t0) * wt * wt;
    acc += fabsf(yp - yt);
  }
  sm[threadIdx.x] = acc;
  __syncthreads();
  for (int sft = 128; sft > 0; sft >>= 1) {
    if ((int)threadIdx.x < sft)
      sm[threadIdx.x] += sm[threadIdx.x + sft];
    __syncthreads();
  }
  if (threadIdx.x == 0) partials[blockIdx.x] = sm[0];
}

__global__ __launch_bounds__(256)
void finalize_kernel(const float* __restrict__ partials, int nb,
                     float* __restrict__ out) {
  __shared__ float sm[256];
  float acc = 0.f;
  for (int i = threadIdx.x; i < nb; i += 256) acc += partials[i];  // fixed order
  sm[threadIdx.x] = acc;
  __syncthreads();
  for (int sft = 128; sft > 0; sft >>= 1) {
    if ((int)threadIdx.x < sft)
      sm[threadIdx.x] += sm[threadIdx.x + sft];
    __syncthreads();
  }
  if (threadIdx.x == 0) out[0] = sm[0] / N_MEAN;
}

// ---------------------------------------------------------------------------
extern "C" void kernel_launch(void* const* d_in, const int* in_sizes, int n_in,
                              void* d_out, int out_size, void* d_ws, size_t ws_size,
                              hipStream_t stream) {
  const float* y_pred   = (const float*)d_in[0];
  const float* y_target = (const float*)d_in[1];
  float* out = (float*)d_out;

  char* ws = (char*)d_ws;
  size_t off = 0;
  float* s_p = (float*)(ws + off); off += (size_t)NIMG * IMG_S * sizeof(float);
  float* s_t = (float*)(ws + off); off += (size_t)NIMG * IMG_S * sizeof(float);
  float* t1  = (float*)(ws + off); off += (size_t)NIMG * IMG_T1 * sizeof(float);
  unsigned* maxslots = (unsigned*)(ws + off); off += 2 * sizeof(unsigned);
  float* partials = (float*)(ws + off); off += 1024 * sizeof(float);
  if (off > ws_size) return;  // ~162.1 MB needed

  const int HB = 7424;  // 64*32*29 waves / 8 waves-per-block
  const int VB = 6728;  // 64*29*29 waves / 8

  init_kernel<<<1, 64, 0, stream>>>(maxslots);
  // pred: horizontal then vertical (t1 reused afterwards)
  hpass_wmma<<<HB, 256, 0, stream>>>(y_pred, t1);
  vpass_wmma<<<VB, 256, 0, stream>>>(t1, s_p);
  // target
  hpass_wmma<<<HB, 256, 0, stream>>>(y_target, t1);
  vpass_wmma<<<VB, 256, 0, stream>>>(t1, s_t);
  // reductions
  max_kernel<<<2048, 256, 0, stream>>>(s_p, s_t, maxslots);
  loss_kernel<<<1024, 256, 0, stream>>>(s_p, s_t, maxslots, partials);
  finalize_kernel<<<1, 256, 0, stream>>>(partials, 1024, out);
}